// FactorizedQuantizedDistribution_41162966565089
// MI455X (gfx1250) — compile-verified
//
#include <hip/hip_runtime.h>
#include <hip/hip_bf16.h>

typedef float v2f __attribute__((ext_vector_type(2)));
typedef float v8f __attribute__((ext_vector_type(8)));

#define HW (256 * 256)
#define BLOCKS_PER_BATCH 64
#define THREADS 256
#define PIX_PER_THREAD 4  // 64*256*4 = 65536 = HW

// Wave-wide (32-lane) f32 sum using V_WMMA_F32_16X16X4_F32.
// A (16x4): lanes 0-15 hold A[m=lane, k=0] in v0, k=1 in v1;
//           lanes 16-31 hold A[m=lane-16, k=2] in v0, k=3 in v1.
// Put per-lane partial p in v0, zero in v1. With B = all ones:
//   D[m,n] = sum_k A[m,k] = p_m + p_{m+16}.
// D layout: lane<16 holds D[r, lane] in reg r (r=0..7 -> M=0..7),
//           lane>=16 holds D[r+8, lane-16]. Summing the 8 D regs gives
//   lanes 0-15:  sum_{m=0..7}(p_m+p_{m+16})
//   lanes 16-31: sum_{m=8..15}(p_m+p_{m+16})
// so q + shfl_xor(q,16) = sum of all 32 partials, in every lane.
__device__ __forceinline__ float wave_sum_wmma(float p) {
    v2f a;
    a.x = p;
    a.y = 0.0f;
    v2f b;
    b.x = 1.0f;
    b.y = 1.0f;
    v8f c = {};
    v8f d = __builtin_amdgcn_wmma_f32_16x16x4_f32(
        /*neg_a=*/false, a, /*neg_b=*/false, b,
        /*c_mod=*/(short)0, c, /*reuse_a=*/false, /*reuse_b=*/false);
    float q = d[0] + d[1] + d[2] + d[3] + d[4] + d[5] + d[6] + d[7];
    q += __shfl_xor(q, 16, 32);
    return q;
}

// Stage 1: grid (BLOCKS_PER_BATCH, B). Each block reduces 1024 pixels of one
// batch image into one partial sum in d_ws.
__global__ void __launch_bounds__(THREADS)
fqd_stage1(const float* __restrict__ params, const float* __restrict__ sample,
           float* __restrict__ partial) {
    const int b = blockIdx.y;
    const int blk = blockIdx.x;
    const int tid = threadIdx.x;

    const float* samp = sample + (size_t)b * HW;
    const float* par = params + (size_t)b * 255 * HW;

    float lp = 0.0f;

#pragma unroll
    for (int i = 0; i < PIX_PER_THREAD; ++i) {
        const int pix = blk * (THREADS * PIX_PER_THREAD) + i * THREADS + tid;
        // bucket id: truncating float->uint8 cast; sample in [0,1)
        int s = (int)(samp[pix] * 256.0f);
        s &= 255;

        const float* pp = par + pix;
        // All 8 tree-node addresses are known up front:
        //   node_x = (2^x - 1) + (s >> (8 - x))
        // Issue the 8 gathers together for MLP.
        float l[8];
#pragma unroll
        for (int x = 0; x < 8; ++x) {
            const int node = ((1 << x) - 1) + (s >> (8 - x));
            l[x] = pp[(size_t)node * HW];
        }
#pragma unroll
        for (int x = 0; x < 8; ++x) {
            const int bit = (s >> (7 - x)) & 1;
            const float y = bit ? -l[x] : l[x];
            // contribution = v*logit - softplus(logit) = -softplus(y)
            // stable softplus: max(y,0) + log1p(exp(-|y|))
            const float sp = fmaxf(y, 0.0f) + log1pf(expf(-fabsf(y)));
            lp -= sp;
        }
    }

    // Intra-wave reduction via WMMA (EXEC is all-1s here: no divergence).
    const float wtot = wave_sum_wmma(lp);

    __shared__ float wsum[THREADS / 32];
    const int wave = tid >> 5;
    const int lane = tid & 31;
    if (lane == 0) wsum[wave] = wtot;
    __syncthreads();

    if (tid == 0) {
        float t = 0.0f;
#pragma unroll
        for (int w = 0; w < THREADS / 32; ++w) t += wsum[w];
        partial[b * BLOCKS_PER_BATCH + blk] = t;
    }
}

// Stage 2: grid (B), block (BLOCKS_PER_BATCH). Deterministic fixed-order tree
// reduction of the 64 block partials per batch.
__global__ void __launch_bounds__(BLOCKS_PER_BATCH)
fqd_stage2(const float* __restrict__ partial, float* __restrict__ out) {
    __shared__ float s[BLOCKS_PER_BATCH];
    const int b = blockIdx.x;
    const int tid = threadIdx.x;
    s[tid] = partial[b * BLOCKS_PER_BATCH + tid];
    __syncthreads();
#pragma unroll
    for (int off = BLOCKS_PER_BATCH / 2; off > 0; off >>= 1) {
        if (tid < off) s[tid] += s[tid + off];
        __syncthreads();
    }
    if (tid == 0) out[b] = s[0];
}

extern "C" void kernel_launch(void* const* d_in, const int* in_sizes, int n_in,
                              void* d_out, int out_size, void* d_ws, size_t ws_size,
                              hipStream_t stream) {
    const float* params = (const float*)d_in[0];  // [8, 255, 256, 256] f32
    const float* sample = (const float*)d_in[1];  // [8, 256, 256]      f32
    float* out = (float*)d_out;                   // [8] f32
    float* partial = (float*)d_ws;                // 8*64 floats = 2 KB

    fqd_stage1<<<dim3(BLOCKS_PER_BATCH, 8), THREADS, 0, stream>>>(params, sample, partial);
    fqd_stage2<<<dim3(8), BLOCKS_PER_BATCH, 0, stream>>>(partial, out);
}